// GraphAttentionLayer_56573309224773
// MI455X (gfx1250) — compile-verified
//
#include <hip/hip_runtime.h>
#include <hip/hip_bf16.h>

typedef float v2f __attribute__((ext_vector_type(2)));
typedef float v8f __attribute__((ext_vector_type(8)));

#define FEAT 512
#define ALPHA 0.1f

// ---------------------------------------------------------------------------
// K1: v[k] = dot(W row k, a2);  s0 = sum_k h[k] * dot(W row k, a1)
// One block, 512 threads = 16 waves, wave-per-row, contiguous row reads.
// ---------------------------------------------------------------------------
__global__ void k_prep(const float* __restrict__ h, const float* __restrict__ W,
                       const float* __restrict__ a, float* __restrict__ v,
                       float* __restrict__ s0out) {
    __shared__ float s0buf[FEAT];
    int tid  = threadIdx.x;        // 0..511
    int lane = tid & 31;
    int wid  = tid >> 5;           // 0..15
    float a1r[16], a2r[16];
#pragma unroll
    for (int t = 0; t < 16; ++t) {
        a1r[t] = a[t * 32 + lane];
        a2r[t] = a[FEAT + t * 32 + lane];
    }
    for (int r = wid; r < FEAT; r += 16) {
        const float* Wr = W + (size_t)r * FEAT;
        float d1 = 0.0f, d2 = 0.0f;
#pragma unroll
        for (int t = 0; t < 16; ++t) {
            float w = Wr[t * 32 + lane];
            d1 += w * a1r[t];
            d2 += w * a2r[t];
        }
        for (int off = 16; off; off >>= 1) {
            d1 += __shfl_xor(d1, off);
            d2 += __shfl_xor(d2, off);
        }
        if (lane == 0) { v[r] = d2; s0buf[r] = h[r] * d1; }
    }
    __syncthreads();
    for (int s = 256; s > 0; s >>= 1) {
        if (tid < s) s0buf[tid] += s0buf[tid + s];
        __syncthreads();
    }
    if (tid == 0) *s0out = s0buf[0];
}

// ---------------------------------------------------------------------------
// K2: scores[i] = leakyrelu(s0 + adj[i,:] . v)   via V_WMMA_F32_16X16X4_F32
// Each wave owns 16 rows. A = 16x4 adj tile (lanes 0-15: K0/K1, 16-31: K2/K3,
// per ISA A layout). B column 0 = v[kb..kb+4), other columns zero
// (B VGPR0 = K0/K2 half-split, VGPR1 = K1/K3). C column 0 = the 16 dots.
// ---------------------------------------------------------------------------
__global__ void k_scores(const float* __restrict__ adj, const float* __restrict__ v,
                         const float* __restrict__ s0p, float* __restrict__ scores,
                         int N) {
    int lane = threadIdx.x & 31;
    int wid  = (blockIdx.x * blockDim.x + threadIdx.x) >> 5;
    int rowbase = wid * 16;
    if (rowbase >= N) return;
    int half = lane >> 4;          // 0 or 1
    int m    = lane & 15;
    int row  = rowbase + m;
    if (row >= N) row = N - 1;     // clamp: OOB rows computed then discarded
    const float* rp = adj + (size_t)row * FEAT + half * 2;

    float vr[16];                  // register-resident copy of v (32 lanes x 16)
#pragma unroll
    for (int t = 0; t < 16; ++t) vr[t] = v[t * 32 + lane];

    v8f c = {0.f, 0.f, 0.f, 0.f, 0.f, 0.f, 0.f, 0.f};
    for (int t = 0; t < 16; ++t) {
#pragma unroll
        for (int s8 = 0; s8 < 8; ++s8) {
            int kb = t * 32 + s8 * 4;
            float2 av = *(const float2*)(rp + kb);      // A: (K+2*half, K+2*half+1)
            v2f A; A.x = av.x; A.y = av.y;
            int i0 = s8 * 4 + half * 2;
            float b0 = __shfl(vr[t], i0);
            float b1 = __shfl(vr[t], i0 + 1);
            v2f B; B.x = (m == 0) ? b0 : 0.0f;          // only column N=0 carries v
                   B.y = (m == 0) ? b1 : 0.0f;
            c = __builtin_amdgcn_wmma_f32_16x16x4_f32(false, A, false, B,
                                                      (short)0, c, false, false);
        }
    }
    float s0 = *s0p;
    if (m == 0) {                  // N=0 column lives in lanes 0 and 16
#pragma unroll
        for (int r = 0; r < 8; ++r) {
            int ro = rowbase + r + 8 * half;            // C VGPRr: M=r / M=r+8
            if (ro < N) {
                float sc = c[r] + s0;
                scores[ro] = (sc >= 0.0f) ? sc : ALPHA * sc;
            }
        }
    }
}

// ---------------------------------------------------------------------------
// K3: softmax over scores, in place (scores -> attention). One block,
// deterministic tree reductions.
// ---------------------------------------------------------------------------
__global__ void k_softmax(float* __restrict__ scores, int N) {
    __shared__ float red[1024];
    __shared__ float s_max, s_inv;
    int t = threadIdx.x;
    float mx = -3.0e38f;
    for (int i = t; i < N; i += 1024) mx = fmaxf(mx, scores[i]);
    red[t] = mx; __syncthreads();
    for (int s = 512; s > 0; s >>= 1) {
        if (t < s) red[t] = fmaxf(red[t], red[t + s]);
        __syncthreads();
    }
    if (t == 0) s_max = red[0];
    __syncthreads();
    mx = s_max;
    float sum = 0.0f;
    for (int i = t; i < N; i += 1024) sum += expf(scores[i] - mx);
    red[t] = sum; __syncthreads();
    for (int s = 512; s > 0; s >>= 1) {
        if (t < s) red[t] += red[t + s];
        __syncthreads();
    }
    if (t == 0) s_inv = 1.0f / red[0];
    __syncthreads();
    float inv = s_inv;
    for (int i = t; i < N; i += 1024) scores[i] = expf(scores[i] - mx) * inv;
}

// ---------------------------------------------------------------------------
// K4: partial_u[chunk, :] = sum_{i in chunk} att[i] * adj[i, :]  via WMMA.
// C = A x B with A row 0 = att chunk (other rows zero), B = 4x16 adj tile.
// B loads are 64B-contiguous per lane-half -> fully coalesced streaming.
// One wave per (column tile, row chunk); 32 tiles x NCHUNK chunks of 512 rows.
// ---------------------------------------------------------------------------
__global__ void k_accum(const float* __restrict__ adj, const float* __restrict__ att,
                        float* __restrict__ partial, int N, int NCHUNK) {
    int lane  = threadIdx.x & 31;
    int g     = blockIdx.x * (blockDim.x >> 5) + (threadIdx.x >> 5);
    int tile  = g & 31;
    int chunk = g >> 5;
    if (chunk >= NCHUNK) return;
    int half = lane >> 4;
    int m    = lane & 15;
    int cb   = tile * 16;

    v8f c = {0.f, 0.f, 0.f, 0.f, 0.f, 0.f, 0.f, 0.f};
    int base = chunk * 512;
    for (int g32 = 0; g32 < 16; ++g32) {
        int ib = base + g32 * 32;
        if (ib >= N) break;                       // uniform per wave
        int ai = ib + lane;
        float attv = (ai < N) ? att[ai] : 0.0f;   // coalesced 32-wide att chunk
#pragma unroll
        for (int s8 = 0; s8 < 8; ++s8) {
            int i  = ib + s8 * 4;
            int r0 = i + half * 2;                // B VGPR0 row (K0/K2 split)
            float b0 = (r0     < N) ? adj[(size_t)r0       * FEAT + cb + m] : 0.0f;
            float b1 = (r0 + 1 < N) ? adj[(size_t)(r0 + 1) * FEAT + cb + m] : 0.0f;
            int i0 = s8 * 4 + half * 2;
            float a0 = __shfl(attv, i0);
            float a1 = __shfl(attv, i0 + 1);
            v2f A; A.x = (m == 0) ? a0 : 0.0f;    // only A row M=0 carries att
                   A.y = (m == 0) ? a1 : 0.0f;
            v2f B; B.x = b0; B.y = b1;
            c = __builtin_amdgcn_wmma_f32_16x16x4_f32(false, A, false, B,
                                                      (short)0, c, false, false);
        }
    }
    // C row 0 (VGPR0, lanes 0-15) holds u-partials for this column tile
    if (lane < 16) partial[(size_t)chunk * FEAT + cb + lane] = c[0];
}

// ---------------------------------------------------------------------------
// K5: u = sum of partials; out[j] = sum_k u[k] * W1[k, j]. One block.
// ---------------------------------------------------------------------------
__global__ void k_finish(const float* __restrict__ partial, const float* __restrict__ W1,
                         float* __restrict__ out, int NCHUNK) {
    __shared__ float u[FEAT];
    int j = threadIdx.x;           // 512 threads
    float s = 0.0f;
    for (int ch = 0; ch < NCHUNK; ++ch) s += partial[(size_t)ch * FEAT + j];
    u[j] = s;
    __syncthreads();
    float o = 0.0f;
    for (int k = 0; k < FEAT; ++k) o += u[k] * W1[(size_t)k * FEAT + j];
    out[j] = o;
}

extern "C" void kernel_launch(void* const* d_in, const int* in_sizes, int n_in,
                              void* d_out, int out_size, void* d_ws, size_t ws_size,
                              hipStream_t stream) {
    (void)n_in; (void)out_size; (void)ws_size;
    const float* h   = (const float*)d_in[0];
    const float* adj = (const float*)d_in[1];
    const float* W   = (const float*)d_in[2];
    const float* a   = (const float*)d_in[3];
    const float* W1  = (const float*)d_in[4];
    float* out = (float*)d_out;
    int N = in_sizes[1] / FEAT;

    // workspace layout (floats): [0]=s0, [16..528)=v, [1024..1024+N)=scores/att,
    // then NCHUNK x 512 partial u vectors.
    float* wsf     = (float*)d_ws;
    float* s0      = wsf;
    float* v       = wsf + 16;
    float* scores  = wsf + 1024;
    int    npad    = (N + 255) & ~255;
    float* partial = wsf + 1024 + npad;
    int NCHUNK = (N + 511) / 512;

    k_prep<<<1, 512, 0, stream>>>(h, W, a, v, s0);

    int nwaves = (N + 15) / 16;
    int nblk   = (nwaves + 7) / 8;
    k_scores<<<nblk, 256, 0, stream>>>(adj, v, s0, scores, N);

    k_softmax<<<1, 1024, 0, stream>>>(scores, N);

    k_accum<<<NCHUNK * 4, 256, 0, stream>>>(adj, scores, partial, N, NCHUNK);

    k_finish<<<1, 512, 0, stream>>>(partial, W1, out, NCHUNK);
}